// NN_log_den_isotropic_gauss_27281632264837
// MI455X (gfx1250) — compile-verified
//
#include <hip/hip_runtime.h>

// CDNA5 / gfx1250, wave32.
// Memory-bound row-wise squared-distance reduction:
//   out[i] = -sum_c (X[i][c] - mean[c])^2 / scale
// 512 MB of X streamed once via global_load_b128 (~22us floor @ 23.3 TB/s);
// per-row horizontal sum done with V_WMMA_F32_16X16X4_F32 (B = ones =>
// D[m][*] = sum_k A[m][k]), exact f32. Two independent accumulators per lane
// are fed as A's K=0/K=1 slots so the WMMA also performs their final add,
// halving the fmac dependency chain seen in round-1 disassembly.

typedef __attribute__((ext_vector_type(2))) float v2f;
typedef __attribute__((ext_vector_type(8))) float v8f;

#define WAVES_PER_BLOCK 8
#define BLOCK_THREADS   256   // 8 wave32 waves

__global__ __launch_bounds__(BLOCK_THREADS)
void gauss_logden_wmma(const float* __restrict__ X,
                       const float* __restrict__ mean,
                       const float* __restrict__ scale,
                       float* __restrict__ out,
                       int num_tiles)
{
    const int lane   = threadIdx.x & 31;
    const int waveId = blockIdx.x * WAVES_PER_BLOCK + (threadIdx.x >> 5);
    const int nWaves = gridDim.x * WAVES_PER_BLOCK;
    const int half   = lane >> 4;   // 0: cols 0..31, 1: cols 32..63
    const int r      = lane & 15;   // row within the 16-row tile

    // Hoist this lane's 32 mean values into registers (mean is only 256 B).
    const float4* M4 = (const float4*)(mean + half * 32);
    float4 m[8];
#pragma unroll
    for (int j = 0; j < 8; ++j) m[j] = M4[j];

    const float neg_inv_scale = -1.0f / scale[0];

    // B matrix (4x16 f32) = all ones: D[m][n] = sum_k A[m][k].
    v2f b; b.x = 1.0f; b.y = 1.0f;

    for (int t = waveId; t < num_tiles; t += nWaves) {
        // Prefetch next tile's slab for this lane (wave-uniform condition).
        const int tn = t + nWaves;
        if (tn < num_tiles) {
            __builtin_prefetch(X + ((size_t)tn * 16 + r) * 64 + half * 32, 0, 0);
        }

        const size_t row = (size_t)t * 16 + r;
        const float4* X4 = (const float4*)(X + row * 64 + half * 32);

        float4 x[8];
#pragma unroll
        for (int j = 0; j < 8; ++j) x[j] = X4[j];   // 8x global_load_b128

        // Two independent accumulator chains (even/odd float4), merged by WMMA.
        float acc0 = 0.0f, acc1 = 0.0f;
#pragma unroll
        for (int j = 0; j < 8; j += 2) {
            float a0 = x[j].x - m[j].x;       acc0 = fmaf(a0, a0, acc0);
            float b0 = x[j + 1].x - m[j + 1].x; acc1 = fmaf(b0, b0, acc1);
            float a1 = x[j].y - m[j].y;       acc0 = fmaf(a1, a1, acc0);
            float b1 = x[j + 1].y - m[j + 1].y; acc1 = fmaf(b1, b1, acc1);
            float a2 = x[j].z - m[j].z;       acc0 = fmaf(a2, a2, acc0);
            float b2 = x[j + 1].z - m[j + 1].z; acc1 = fmaf(b2, b2, acc1);
            float a3 = x[j].w - m[j].w;       acc0 = fmaf(a3, a3, acc0);
            float b3 = x[j + 1].w - m[j + 1].w; acc1 = fmaf(b3, b3, acc1);
        }

        // Pre-scale partials by -1/scale (distributes over the sum), feed WMMA.
        // A layout (16x4 f32): lanes 0-15 hold {K=0,K=1}, lanes 16-31 {K=2,K=3};
        // with B = ones the WMMA computes acc0+acc1 across both lane halves.
        v2f a; a.x = acc0 * neg_inv_scale; a.y = acc1 * neg_inv_scale;
        v8f c = {};
        c = __builtin_amdgcn_wmma_f32_16x16x4_f32(
                /*neg_a=*/false, a, /*neg_b=*/false, b,
                /*c_mod=*/(short)0, c, /*reuse_a=*/false, /*reuse_b=*/false);

        // D layout (16x16 f32): VGPR v, lanes 0-15 -> M=v; lanes 16-31 -> M=v+8.
        // Every column of D is identical, so lane r==0 of each half stores 8 rows.
        if (r == 0) {
            float4 o0, o1;
            o0.x = c[0]; o0.y = c[1]; o0.z = c[2]; o0.w = c[3];
            o1.x = c[4]; o1.y = c[5]; o1.z = c[6]; o1.w = c[7];
            float4* O = (float4*)(out + (size_t)t * 16 + half * 8);
            O[0] = o0;   // global_store_b128
            O[1] = o1;   // global_store_b128
        }
    }
}

// Scalar tail for N % 16 != 0 (not hit for N = 2,000,000, kept for determinism).
__global__ void gauss_logden_tail(const float* __restrict__ X,
                                  const float* __restrict__ mean,
                                  const float* __restrict__ scale,
                                  float* __restrict__ out,
                                  int start_row, int n_rows)
{
    int i = blockIdx.x * blockDim.x + threadIdx.x;
    if (i >= n_rows) return;
    size_t row = (size_t)(start_row + i);
    float acc = 0.0f;
    for (int c = 0; c < 64; ++c) {
        float d = X[row * 64 + c] - mean[c];
        acc = fmaf(d, d, acc);
    }
    out[row] = -acc / scale[0];
}

extern "C" void kernel_launch(void* const* d_in, const int* in_sizes, int n_in,
                              void* d_out, int out_size, void* d_ws, size_t ws_size,
                              hipStream_t stream)
{
    const float* X     = (const float*)d_in[0];  // [N, 64] f32
    const float* mean  = (const float*)d_in[1];  // [1, 64] f32
    const float* scale = (const float*)d_in[2];  // [1]     f32
    float*       out   = (float*)d_out;          // [N]     f32

    const long N = (long)in_sizes[0] / 64;
    const int  num_tiles = (int)(N / 16);

    if (num_tiles > 0) {
        // ~4 tiles per wave; cap grid, 8 waves per block.
        int waves_needed = (num_tiles + 3) / 4;
        int blocks = (waves_needed + WAVES_PER_BLOCK - 1) / WAVES_PER_BLOCK;
        if (blocks > 4096) blocks = 4096;
        if (blocks < 1)    blocks = 1;
        gauss_logden_wmma<<<blocks, BLOCK_THREADS, 0, stream>>>(
            X, mean, scale, out, num_tiles);
    }

    const int tail = (int)(N - (long)num_tiles * 16);
    if (tail > 0) {
        gauss_logden_tail<<<1, 32, 0, stream>>>(
            X, mean, scale, out, num_tiles * 16, tail);
    }
}